// FloodGNNGRU_30906584662493
// MI455X (gfx1250) — compile-verified
//
#include <hip/hip_runtime.h>
#include <hip/hip_bf16.h>

// ---------------------------------------------------------------------------
// MI455X (gfx1250, wave32) FloodGNN-GRU forward pass.
// All dense GVP matmuls run on V_WMMA_F32_16X16X4_F32; one wave owns a 16-row
// tile (16 nodes or 16 edges). Operand tiles are staged through LDS with b128
// loads and clamped (branch-free) addressing; edge messages are fused
// gather -> GVP (WMMA) -> atomicAdd scatter.
// ---------------------------------------------------------------------------

typedef float v2f __attribute__((ext_vector_type(2)));
typedef float v8f __attribute__((ext_vector_type(8)));

__device__ __forceinline__ v8f wmma4(v2f a, v2f b, v8f c) {
  // D = A(16x4 f32) * B(4x16 f32) + C(16x16 f32)
  return __builtin_amdgcn_wmma_f32_16x16x4_f32(false, a, false, b, (short)0, c,
                                               false, false);
}
static __device__ __forceinline__ v8f zero8() {
  v8f z = {0.f, 0.f, 0.f, 0.f, 0.f, 0.f, 0.f, 0.f};
  return z;
}
static __device__ __forceinline__ float sigf(float x) {
  return 1.f / (1.f + expf(-x));
}
// order-preserving float <-> uint mapping for single-instruction atomic max
static __device__ __forceinline__ unsigned fkey(float f) {
  const unsigned u = __float_as_uint(f);
  return (u & 0x80000000u) ? ~u : (u | 0x80000000u);
}
static __device__ __forceinline__ float funkey(unsigned k) {
  return __uint_as_float((k & 0x80000000u) ? (k & 0x7fffffffu) : ~k);
}

// ---------------------------------------------------------------------------
// Generic per-node GVP:  vh = v @ Wh,  s_out = act([s ‖ |vh|] @ Ws + bs),
//                        v_out = act(vh @ Wv).  One wave per 16-node tile.
// H == 16 for every GVP in this model.
// ---------------------------------------------------------------------------
template <int SI, int VI, int SO, int VO, bool SACT, bool VACT, bool ACC>
__global__ __launch_bounds__(32) void gvp_node_kernel(
    const float* __restrict__ Sin, const float* __restrict__ Vin,
    const float* __restrict__ Wh, const float* __restrict__ Ws,
    const float* __restrict__ Wv, const float* __restrict__ Bs,
    float* __restrict__ Sout, float* __restrict__ Vout, int n) {
  constexpr int SK = SI + 16;            // concat width
  constexpr int SKP = (SK + 3) & ~3;     // padded to K-step multiple
  constexpr int NT = (SO + 15) / 16;
  const int lane = threadIdx.x;
  const int arow = lane & 15;
  const int ahalf = lane >> 4;
  const int ncol = arow;
  const int mbase = 8 * ahalf;
  const int row0 = blockIdx.x * 16;
  const int nodeA = row0 + arow;
  const bool rok = nodeA < n;
  const int nodeC = rok ? nodeA : (n - 1);
  const float rsc = rok ? 1.f : 0.f;

  __shared__ float sA[16][SKP];       // [s | vn | pad] per row
  __shared__ float vhs[16][2][16];    // vh, d-major for contiguous K reads

  // ---- stage s-tile into LDS (clamped rows, OOB rows zeroed) ----
  if constexpr (SI % 4 == 0) {
    constexpr int NQ = 4 * SI;  // float4 chunks in tile
#pragma unroll
    for (int q = 0; q < NQ; q += 32) {
      const int qq = q + lane;
      const int row = qq / (SI / 4);
      const int c4 = qq % (SI / 4);
      const int node = row0 + row;
      const int nc = (node < n) ? node : (n - 1);
      const float sc = (node < n) ? 1.f : 0.f;
      const float4 x = *(const float4*)(Sin + (size_t)nc * SI + c4 * 4);
      float4 y;
      y.x = x.x * sc;
      y.y = x.y * sc;
      y.z = x.z * sc;
      y.w = x.w * sc;
      *(float4*)&sA[row][c4 * 4] = y;
    }
  } else {
    constexpr int NE = 16 * SI;
#pragma unroll
    for (int q = 0; q < NE; q += 32) {
      const int qq = q + lane;
      if (qq < NE) {
        const int row = qq / SI, c = qq % SI;
        const int node = row0 + row;
        const int nc = (node < n) ? node : (n - 1);
        const float sc = (node < n) ? 1.f : 0.f;
        sA[row][c] = Sin[(size_t)nc * SI + c] * sc;
      }
    }
  }
  __syncthreads();

  // ---- vh = einsum('nvd,vh->nhd') : M=16 x K=VI x N=16, per d ----
  v8f av0 = zero8(), av1 = zero8();
#pragma unroll
  for (int kb = 0; kb < VI; kb += 4) {
    const int k0 = kb + 2 * ahalf;
    const bool k0ok = k0 < VI, k1ok = (k0 + 1) < VI;
    v2f a0, a1, b;
    a0.x = k0ok ? Vin[((size_t)nodeC * VI + k0) * 2 + 0] * rsc : 0.f;
    a0.y = k1ok ? Vin[((size_t)nodeC * VI + k0 + 1) * 2 + 0] * rsc : 0.f;
    a1.x = k0ok ? Vin[((size_t)nodeC * VI + k0) * 2 + 1] * rsc : 0.f;
    a1.y = k1ok ? Vin[((size_t)nodeC * VI + k0 + 1) * 2 + 1] * rsc : 0.f;
    b.x = k0ok ? Wh[k0 * 16 + ncol] : 0.f;
    b.y = k1ok ? Wh[(k0 + 1) * 16 + ncol] : 0.f;
    av0 = wmma4(a0, b, av0);
    av1 = wmma4(a1, b, av1);
  }
#pragma unroll
  for (int r = 0; r < 8; ++r) {
    const int m = mbase + r;
    const float x0 = av0[r], x1 = av1[r];
    vhs[m][0][ncol] = x0;
    vhs[m][1][ncol] = x1;
    sA[m][SI + ncol] = sqrtf(fmaxf(x0 * x0 + x1 * x1, 1e-8f));
    if constexpr (SKP > SK) {
      if (ncol < SKP - SK) sA[m][SK + ncol] = 0.f;
    }
  }
  __syncthreads();

  // ---- s_out = [s ‖ vn] @ Ws + bs ----
  v8f as[NT];
#pragma unroll
  for (int t = 0; t < NT; ++t) as[t] = zero8();
#pragma unroll
  for (int kb = 0; kb < SK; kb += 4) {
    const int k0 = kb + 2 * ahalf;
    const int k1 = k0 + 1;
    v2f a;
    a.x = sA[arow][k0];  // pad columns are zero
    a.y = sA[arow][k1];
#pragma unroll
    for (int t = 0; t < NT; ++t) {
      const int c = t * 16 + ncol;
      v2f b;
      b.x = (k0 < SK && c < SO) ? Ws[k0 * SO + c] : 0.f;
      b.y = (k1 < SK && c < SO) ? Ws[k1 * SO + c] : 0.f;
      as[t] = wmma4(a, b, as[t]);
    }
  }
#pragma unroll
  for (int t = 0; t < NT; ++t)
#pragma unroll
    for (int r = 0; r < 8; ++r) {
      const int m = mbase + r, c = t * 16 + ncol, node = row0 + m;
      if (c < SO && node < n) {
        float x = as[t][r] + Bs[c];
        if (SACT) x = fmaxf(x, 0.f);
        if (ACC)
          Sout[(size_t)node * SO + c] += x;
        else
          Sout[(size_t)node * SO + c] = x;
      }
    }

  // ---- v_out = vh @ Wv (per d), optional gated-norm activation ----
  v8f aw0 = zero8(), aw1 = zero8();
#pragma unroll
  for (int kb = 0; kb < 16; kb += 4) {
    const int k0 = kb + 2 * ahalf;
    v2f a0, a1, b;
    a0.x = vhs[arow][0][k0];
    a0.y = vhs[arow][0][k0 + 1];
    a1.x = vhs[arow][1][k0];
    a1.y = vhs[arow][1][k0 + 1];
    b.x = (ncol < VO) ? Wv[k0 * VO + ncol] : 0.f;
    b.y = (ncol < VO) ? Wv[(k0 + 1) * VO + ncol] : 0.f;
    aw0 = wmma4(a0, b, aw0);
    aw1 = wmma4(a1, b, aw1);
  }
#pragma unroll
  for (int r = 0; r < 8; ++r) {
    const int m = mbase + r, node = row0 + m;
    float x0 = aw0[r], x1 = aw1[r];
    if (VACT) {
      const float nv = sqrtf(fmaxf(x0 * x0 + x1 * x1, 1e-8f));
      const float sg = sigf(nv);
      x0 *= sg;
      x1 *= sg;
    }
    if (ncol < VO && node < n) {
      if (ACC) {
        Vout[((size_t)node * VO + ncol) * 2 + 0] += x0;
        Vout[((size_t)node * VO + ncol) * 2 + 1] += x1;
      } else {
        Vout[((size_t)node * VO + ncol) * 2 + 0] = x0;
        Vout[((size_t)node * VO + ncol) * 2 + 1] = x1;
      }
    }
  }
}

// ---------------------------------------------------------------------------
// Fused per-edge message GVP (m_gvp; 64,16 -> 64,16 both activations):
// softmax-scaled gather (staged via LDS, b128) -> WMMA GVP -> atomicAdd
// scatter to destination aggregates. One wave per 16-edge tile.
// ---------------------------------------------------------------------------
__global__ __launch_bounds__(32) void edge_message_kernel(
    const int* __restrict__ ei, int E, int Etot, const float* __restrict__ Sne,
    const float* __restrict__ Vne, const float* __restrict__ e_s,
    const float* __restrict__ z_s, const float* __restrict__ e_v,
    const float* __restrict__ z_v, const float* __restrict__ Wh,
    const float* __restrict__ Ws, const float* __restrict__ Wv,
    const float* __restrict__ Bs, float* __restrict__ Sagg,
    float* __restrict__ Vagg) {
  const int lane = threadIdx.x;
  const int arow = lane & 15, ahalf = lane >> 4;
  const int ncol = arow, mbase = 8 * ahalf;
  const int e0 = blockIdx.x * 16;

  __shared__ int sh_src[16], sh_dst[16];
  __shared__ float sh_ws[16], sh_wv[16];
  __shared__ float sA[16][80];      // [ws*s_j | vn]
  __shared__ float vA[16][32];      // wv*v_j (vc,d interleaved)
  __shared__ float vhs[16][2][16];  // vh, d-major

  if (lane < 16) {
    const int e = e0 + lane;
    int s = 0, d = -1;
    float ws = 0.f, wv = 0.f;
    if (e < Etot) {
      s = (e < E) ? ei[e] : (e - E);
      d = (e < E) ? ei[E + e] : (e - E);
      ws = e_s[e] / (z_s[d] + 1e-16f);
      wv = e_v[e] / (z_v[d] + 1e-16f);
    }
    sh_src[lane] = s;  // clamped (0 for padding rows; weight 0 kills them)
    sh_dst[lane] = d;
    sh_ws[lane] = ws;
    sh_wv[lane] = wv;
  }
  __syncthreads();

  // stage ws * s_j rows (16 x 64) as float4
#pragma unroll
  for (int q = 0; q < 256; q += 32) {
    const int qq = q + lane;
    const int row = qq >> 4, c4 = qq & 15;
    const float w = sh_ws[row];
    const float4 x = *(const float4*)(Sne + (size_t)sh_src[row] * 64 + c4 * 4);
    float4 y;
    y.x = x.x * w;
    y.y = x.y * w;
    y.z = x.z * w;
    y.w = x.w * w;
    *(float4*)&sA[row][c4 * 4] = y;
  }
  // stage wv * v_j rows (16 x 32) as float4
#pragma unroll
  for (int q = 0; q < 128; q += 32) {
    const int qq = q + lane;
    const int row = qq >> 3, c4 = qq & 7;
    const float w = sh_wv[row];
    const float4 x = *(const float4*)(Vne + (size_t)sh_src[row] * 32 + c4 * 4);
    float4 y;
    y.x = x.x * w;
    y.y = x.y * w;
    y.z = x.z * w;
    y.w = x.w * w;
    *(float4*)&vA[row][c4 * 4] = y;
  }
  __syncthreads();

  // vh = (wv*v_j) @ Wh
  v8f av0 = zero8(), av1 = zero8();
#pragma unroll
  for (int kb = 0; kb < 16; kb += 4) {
    const int k0 = kb + 2 * ahalf;
    v2f a0, a1, b;
    a0.x = vA[arow][k0 * 2 + 0];
    a0.y = vA[arow][(k0 + 1) * 2 + 0];
    a1.x = vA[arow][k0 * 2 + 1];
    a1.y = vA[arow][(k0 + 1) * 2 + 1];
    b.x = Wh[k0 * 16 + ncol];
    b.y = Wh[(k0 + 1) * 16 + ncol];
    av0 = wmma4(a0, b, av0);
    av1 = wmma4(a1, b, av1);
  }
#pragma unroll
  for (int r = 0; r < 8; ++r) {
    const int m = mbase + r;
    const float x0 = av0[r], x1 = av1[r];
    vhs[m][0][ncol] = x0;
    vhs[m][1][ncol] = x1;
    sA[m][64 + ncol] = sqrtf(fmaxf(x0 * x0 + x1 * x1, 1e-8f));
  }
  __syncthreads();

  // s_m = relu([ws*s_j ‖ vn] @ Ws + bs), scattered to dst
  v8f as[4];
#pragma unroll
  for (int t = 0; t < 4; ++t) as[t] = zero8();
#pragma unroll
  for (int kb = 0; kb < 80; kb += 4) {
    const int k0 = kb + 2 * ahalf;
    const int k1 = k0 + 1;
    v2f a;
    a.x = sA[arow][k0];
    a.y = sA[arow][k1];
#pragma unroll
    for (int t = 0; t < 4; ++t) {
      v2f b;
      b.x = Ws[k0 * 64 + t * 16 + ncol];
      b.y = Ws[k1 * 64 + t * 16 + ncol];
      as[t] = wmma4(a, b, as[t]);
    }
  }
#pragma unroll
  for (int t = 0; t < 4; ++t)
#pragma unroll
    for (int r = 0; r < 8; ++r) {
      const int m = mbase + r, c = t * 16 + ncol;
      const int dd = sh_dst[m];
      if (dd >= 0) {
        const float x = fmaxf(as[t][r] + Bs[c], 0.f);
        atomicAdd(&Sagg[(size_t)dd * 64 + c], x);
      }
    }

  // v_m = act(vh @ Wv), scattered to dst
  v8f aw0 = zero8(), aw1 = zero8();
#pragma unroll
  for (int kb = 0; kb < 16; kb += 4) {
    const int k0 = kb + 2 * ahalf;
    v2f a0, a1, b;
    a0.x = vhs[arow][0][k0];
    a0.y = vhs[arow][0][k0 + 1];
    a1.x = vhs[arow][1][k0];
    a1.y = vhs[arow][1][k0 + 1];
    b.x = Wv[k0 * 16 + ncol];
    b.y = Wv[(k0 + 1) * 16 + ncol];
    aw0 = wmma4(a0, b, aw0);
    aw1 = wmma4(a1, b, aw1);
  }
#pragma unroll
  for (int r = 0; r < 8; ++r) {
    const int m = mbase + r;
    const int dd = sh_dst[m];
    float x0 = aw0[r], x1 = aw1[r];
    const float nv = sqrtf(fmaxf(x0 * x0 + x1 * x1, 1e-8f));
    const float sg = sigf(nv);
    x0 *= sg;
    x1 *= sg;
    if (dd >= 0) {
      atomicAdd(&Vagg[((size_t)dd * 16 + ncol) * 2 + 0], x0);
      atomicAdd(&Vagg[((size_t)dd * 16 + ncol) * 2 + 1], x1);
    }
  }
}

// ---------------------------------------------------------------------------
// Edge attention passes. Self-loops are edges e >= E with src=dst=e-E.
// Segment max uses an order-preserving uint key -> single atomicMax.
// ---------------------------------------------------------------------------
__global__ void att_raw_kernel(const int* __restrict__ ei, int E, int Etot,
                               const float* __restrict__ Sne,
                               const float* __restrict__ Vne,
                               float* __restrict__ att_s,
                               float* __restrict__ att_v,
                               unsigned* __restrict__ m_s,
                               unsigned* __restrict__ m_v) {
  const int e = blockIdx.x * blockDim.x + threadIdx.x;
  if (e >= Etot) return;
  const int s = (e < E) ? ei[e] : (e - E);
  const int d = (e < E) ? ei[E + e] : (e - E);
  const float4* a = (const float4*)(Sne + (size_t)d * 64);
  const float4* b = (const float4*)(Sne + (size_t)s * 64);
  float acc = 0.f;
#pragma unroll
  for (int k = 0; k < 16; ++k) {
    const float4 x = a[k], y = b[k];
    acc += x.x * y.x + x.y * y.y + x.z * y.z + x.w * y.w;
  }
  acc = (acc >= 0.f) ? acc : 0.2f * acc;
  att_s[e] = acc;
  atomicMax(&m_s[d], fkey(acc));

  const float4* c = (const float4*)(Vne + (size_t)d * 32);
  const float4* f = (const float4*)(Vne + (size_t)s * 32);
  float av = 0.f;
#pragma unroll
  for (int k = 0; k < 8; ++k) {
    const float4 x = c[k], y = f[k];
    av += x.x * y.x + x.y * y.y + x.z * y.z + x.w * y.w;
  }
  av = (av >= 0.f) ? av : 0.2f * av;
  att_v[e] = av;
  atomicMax(&m_v[d], fkey(av));
}

__global__ void att_exp_kernel(const int* __restrict__ ei, int E, int Etot,
                               const float* __restrict__ att_s,
                               const float* __restrict__ att_v,
                               const unsigned* __restrict__ m_s,
                               const unsigned* __restrict__ m_v,
                               float* __restrict__ e_s, float* __restrict__ e_v,
                               float* __restrict__ z_s,
                               float* __restrict__ z_v) {
  const int e = blockIdx.x * blockDim.x + threadIdx.x;
  if (e >= Etot) return;
  const int d = (e < E) ? ei[E + e] : (e - E);
  const float es = expf(att_s[e] - funkey(m_s[d]));
  e_s[e] = es;
  atomicAdd(&z_s[d], es);
  const float ev = expf(att_v[e] - funkey(m_v[d]));
  e_v[e] = ev;
  atomicAdd(&z_v[d], ev);
}

__global__ void layer_init_kernel(unsigned* m_s, float* z_s, unsigned* m_v,
                                  float* z_v, float* Sagg, float* Vagg,
                                  int n) {
  const int i = blockIdx.x * blockDim.x + threadIdx.x;
  if (i >= n) return;
  m_s[i] = 0u;  // key below fkey(-inf)
  z_s[i] = 0.f;
  m_v[i] = 0u;
  z_v[i] = 0.f;
  float4 z4 = {0.f, 0.f, 0.f, 0.f};
#pragma unroll
  for (int k = 0; k < 16; ++k) *(float4*)&Sagg[(size_t)i * 64 + k * 4] = z4;
#pragma unroll
  for (int k = 0; k < 8; ++k) *(float4*)&Vagg[(size_t)i * 32 + k * 4] = z4;
}

// ---------------------------------------------------------------------------
// Small elementwise / state kernels
// ---------------------------------------------------------------------------
__global__ void sigmoid_kernel(float* x, int n) {
  const int i = blockIdx.x * blockDim.x + threadIdx.x;
  if (i < n) x[i] = sigf(x[i]);
}
__global__ void tanh_kernel(float* x, int n) {
  const int i = blockIdx.x * blockDim.x + threadIdx.x;
  if (i < n) x[i] = tanhf(x[i]);
}
__global__ void mul_kernel(float* o, const float* a, const float* b, int n) {
  const int i = blockIdx.x * blockDim.x + threadIdx.x;
  if (i < n) o[i] = a[i] * b[i];
}
__global__ void gru_kernel(float* hid, const float* gz, const float* hh,
                           const unsigned char* seq, int tm1, int T, int feats,
                           int n, int has_hid) {
  const int idx = blockIdx.x * blockDim.x + threadIdx.x;
  if (idx >= n * feats) return;
  const int node = idx / feats;
  const float z = gz[idx], h = hh[idx];
  float out = (1.f - z) * h;
  if (has_hid) out += z * hid[idx];
  hid[idx] = (seq[node * T + tm1] != 0) ? out : 0.f;
}

__global__ void build_sin_kernel(const float* s_static, const float* s_h,
                                 const float* r_h, const float* l_h,
                                 float* s_in, int n) {
  const int i = blockIdx.x * blockDim.x + threadIdx.x;
  if (i >= n) return;
  float* o = s_in + (size_t)i * 10;
#pragma unroll
  for (int k = 0; k < 6; ++k) o[k] = s_static[(size_t)i * 6 + k];
  o[6] = s_h[i * 2 + 0];
  o[7] = s_h[i * 2 + 1];
  o[8] = r_h[i];
  o[9] = l_h[i];
}

__global__ void init_state_kernel(const int* binary, const float* x_v,
                                  const float* x_vn, const float* rain,
                                  const float* wdfp, float* s_h, float* v_h,
                                  float* l_h, float* r_h, float* scal,
                                  int* cnt, int n, int T) {
  const int i = blockIdx.x * blockDim.x + threadIdx.x;
  if (i >= n) return;
#pragma unroll
  for (int c = 0; c < 2; ++c) {
    const int b = binary[((size_t)i * T + 0) * 2 + c];
    s_h[i * 2 + c] = b ? x_vn[((size_t)i * T + 0) * 2 + c] : 0.f;
#pragma unroll
    for (int d = 0; d < 2; ++d)
      v_h[((size_t)i * 2 + c) * 2 + d] =
          b ? x_v[(((size_t)i * T + 0) * 2 + c) * 2 + d] : 0.f;
  }
  l_h[i] = wdfp[(size_t)i * T + 0];
  r_h[i] = rain[(size_t)i * T + 0];
  if (i == 0) {
    scal[0] = 0.f;  // ll accumulator
    scal[1] = 0.f;  // fl accumulator
    scal[2] = 0.f;  // total loss
    cnt[0] = 0;     // binary ones count (skip detection)
  }
}

__global__ void feat_epilogue_kernel(
    const float* l_pred, const float* v_old_buf, const unsigned char* seq,
    const int* binary, const float* x_v, const float* x_vn, const float* rain,
    const float* wdfp, float* l_h, float* s_h, float* v_h, float* r_h,
    float* out, float* scal, int* cnt, int ti, int T, int n, int steps,
    int out_size) {
  const int i = blockIdx.x * blockDim.x + threadIdx.x;
  if (i >= n) return;
  const int valid = seq[i * T + (ti - 1)] != 0;
  const float mf = valid ? 1.f : 0.f;
  float l = l_pred[i];
  l = l * l;
  float vo[2][2];
#pragma unroll
  for (int c = 0; c < 2; ++c)
#pragma unroll
    for (int d = 0; d < 2; ++d)
      vo[c][d] = v_old_buf[((size_t)i * 2 + c) * 2 + d];
  const float s0 =
      sqrtf(fmaxf(vo[0][0] * vo[0][0] + vo[0][1] * vo[0][1], 1e-8f));
  const float s1 =
      sqrtf(fmaxf(vo[1][0] * vo[1][0] + vo[1][1] * vo[1][1], 1e-8f));
  const float lm = valid ? l : 0.f;
  l_h[i] = lm;
  s_h[i * 2 + 0] = valid ? s0 : 0.f;
  s_h[i * 2 + 1] = valid ? s1 : 0.f;
  v_h[((size_t)i * 2 + 0) * 2 + 0] = valid ? vo[0][0] / s0 : 0.f;
  v_h[((size_t)i * 2 + 0) * 2 + 1] = valid ? vo[0][1] / s0 : 0.f;
  v_h[((size_t)i * 2 + 1) * 2 + 0] = valid ? vo[1][0] / s1 : 0.f;
  v_h[((size_t)i * 2 + 1) * 2 + 1] = valid ? vo[1][1] / s1 : 0.f;
  const int oidx = i * steps + (ti - 1);
  if (oidx < out_size) out[oidx] = lm;

  const float ll = fabsf(lm - wdfp[(size_t)i * T + ti]) * mf;
  float fl = 0.f;
  int ones = 0;
#pragma unroll
  for (int c = 0; c < 2; ++c) {
    const int b = binary[((size_t)i * T + ti) * 2 + c];
    ones += (b != 0);
    const float vn = b ? x_vn[((size_t)i * T + ti) * 2 + c] : 0.f;
#pragma unroll
    for (int d = 0; d < 2; ++d) {
      const float tv =
          (b ? x_v[(((size_t)i * T + ti) * 2 + c) * 2 + d] : 0.f) * vn;
      fl += fabsf(vo[c][d] - tv);
    }
  }
  fl *= mf;
  atomicAdd(&scal[0], ll);
  atomicAdd(&scal[1], fl);
  atomicAdd(cnt, ones);
  r_h[i] = rain[(size_t)i * T + ti];
}

__global__ void loss_step_kernel(float* scal, int* cnt, int n) {
  const int c = cnt[0];
  const int skip = (c == 0) || (c == 2 * n);
  if (!skip) scal[2] += scal[0] + scal[1];
  scal[0] = 0.f;
  scal[1] = 0.f;
  cnt[0] = 0;
}

__global__ void write_tail_kernel(const float* scal, const unsigned char* seq,
                                  float* out, int n, int T, int steps,
                                  int out_size) {
  const int i = blockIdx.x * blockDim.x + threadIdx.x;
  const int base = n * steps;
  if (i == 0 && base < out_size) out[base] = scal[2];
  if (i < n * T && (base + 1 + i) < out_size)
    out[base + 1 + i] = seq[i] ? 1.f : 0.f;
}

// ---------------------------------------------------------------------------
// Host orchestration
// ---------------------------------------------------------------------------
struct GW {
  const float *Wh, *Ws, *Wv, *Bs;
};

extern "C" void kernel_launch(void* const* d_in, const int* in_sizes, int n_in,
                              void* d_out, int out_size, void* d_ws,
                              size_t ws_size, hipStream_t stream) {
  const int* ei = (const int*)d_in[0];
  const float* s_static = (const float*)d_in[1];
  const unsigned char* seq = (const unsigned char*)d_in[2];
  const int* binary = (const int*)d_in[3];
  const float* x_v = (const float*)d_in[4];
  const float* x_vn = (const float*)d_in[5];
  const float* rain = (const float*)d_in[6];
  const float* wdfp = (const float*)d_in[7];
  // d_in[8] = batch (unused)
  const float* const* P = (const float* const*)(d_in + 9);
  auto gw = [&](int b) { return GW{P[b], P[b + 1], P[b + 2], P[b + 3]}; };
  // pytree (sorted-key) leaf order, 4 leaves per GVP (Wh, Ws, Wv, bs):
  const GW g1 = gw(0), g2 = gw(4), enc = gw(8);
  struct Layer {
    GW m, n, u;
  };
  auto layer = [&](int b) { return Layer{gw(b), gw(b + 4), gw(b + 8)}; };
  const Layer L_hh_h = layer(12), L_hh_in = layer(24), L_r_h = layer(36),
              L_r_in = layer(48), L_z_h = layer(60), L_z_in = layer(72);

  const int N = in_sizes[1] / 6;
  const int E = in_sizes[0] / 2;
  const int Etot = E + N;
  const int T = in_sizes[6] / N;
  const int steps = T - 1;

  // workspace carve-up (16B-aligned chunks)
  float* w = (float*)d_ws;
  auto alloc = [&](size_t cnt) {
    float* p = w;
    w += (cnt + 3) & ~(size_t)3;
    return p;
  };
  float* s_h = alloc((size_t)N * 2);
  float* l_h = alloc(N);
  float* r_h = alloc(N);
  float* v_h = alloc((size_t)N * 4);
  float* s_hid = alloc((size_t)N * 64);
  float* v_hid = alloc((size_t)N * 32);
  float* s_in = alloc((size_t)N * 10);
  float* s_enc = alloc((size_t)N * 64);
  float* v_enc = alloc((size_t)N * 32);
  float* s_ne = alloc((size_t)N * 64);
  float* v_ne = alloc((size_t)N * 32);
  float* s_agg = alloc((size_t)N * 64);
  float* v_agg = alloc((size_t)N * 32);
  float* s_gz = alloc((size_t)N * 64);
  float* v_gz = alloc((size_t)N * 32);
  float* s_gr = alloc((size_t)N * 64);
  float* v_gr = alloc((size_t)N * 32);
  float* s_hh = alloc((size_t)N * 64);
  float* v_hh = alloc((size_t)N * 32);
  float* s_rt = alloc((size_t)N * 64);
  float* v_rt = alloc((size_t)N * 32);
  float* s_f = alloc((size_t)N * 64);
  float* v_f = alloc((size_t)N * 32);
  float* l_pred = alloc(N);
  float* v_old = alloc((size_t)N * 4);
  unsigned* m_s = (unsigned*)alloc(N);
  float* z_s = alloc(N);
  unsigned* m_v = (unsigned*)alloc(N);
  float* z_v = alloc(N);
  float* att_s = alloc(Etot);
  float* att_v = alloc(Etot);
  float* e_s = alloc(Etot);
  float* e_v = alloc(Etot);
  float* scal = alloc(4);
  int* cnt = (int*)alloc(4);
  (void)ws_size;
  (void)n_in;

  const int thr = 256;
  const int nb_n = (N + thr - 1) / thr;
  const int nb_e = (Etot + thr - 1) / thr;
  const int nt16 = (N + 15) / 16;
  const int et16 = (Etot + 15) / 16;

  auto flood_layer = [&](const Layer& L, const float* si, const float* vi,
                         float* so, float* vo, bool acc) {
    gvp_node_kernel<64, 16, 64, 16, true, true, false>
        <<<nt16, 32, 0, stream>>>(si, vi, L.n.Wh, L.n.Ws, L.n.Wv, L.n.Bs, s_ne,
                                  v_ne, N);
    layer_init_kernel<<<nb_n, thr, 0, stream>>>(m_s, z_s, m_v, z_v, s_agg,
                                                v_agg, N);
    att_raw_kernel<<<nb_e, thr, 0, stream>>>(ei, E, Etot, s_ne, v_ne, att_s,
                                             att_v, m_s, m_v);
    att_exp_kernel<<<nb_e, thr, 0, stream>>>(ei, E, Etot, att_s, att_v, m_s,
                                             m_v, e_s, e_v, z_s, z_v);
    edge_message_kernel<<<et16, 32, 0, stream>>>(
        ei, E, Etot, s_ne, v_ne, e_s, z_s, e_v, z_v, L.m.Wh, L.m.Ws, L.m.Wv,
        L.m.Bs, s_agg, v_agg);
    if (acc)
      gvp_node_kernel<64, 16, 64, 16, true, true, true>
          <<<nt16, 32, 0, stream>>>(s_agg, v_agg, L.u.Wh, L.u.Ws, L.u.Wv,
                                    L.u.Bs, so, vo, N);
    else
      gvp_node_kernel<64, 16, 64, 16, true, true, false>
          <<<nt16, 32, 0, stream>>>(s_agg, v_agg, L.u.Wh, L.u.Ws, L.u.Wv,
                                    L.u.Bs, so, vo, N);
  };

  init_state_kernel<<<nb_n, thr, 0, stream>>>(binary, x_v, x_vn, rain, wdfp,
                                              s_h, v_h, l_h, r_h, scal, cnt, N,
                                              T);

  for (int i = 1; i < T; ++i) {
    const int hid = (i > 1);
    build_sin_kernel<<<nb_n, thr, 0, stream>>>(s_static, s_h, r_h, l_h, s_in,
                                               N);
    // enc GVP (no activations): (s_in 10, v_h 2) -> (64, 16)
    gvp_node_kernel<10, 2, 64, 16, false, false, false>
        <<<nt16, 32, 0, stream>>>(s_in, v_h, enc.Wh, enc.Ws, enc.Wv, enc.Bs,
                                  s_enc, v_enc, N);
    flood_layer(L_z_in, s_enc, v_enc, s_gz, v_gz, false);
    flood_layer(L_r_in, s_enc, v_enc, s_gr, v_gr, false);
    if (hid) {
      flood_layer(L_z_h, s_hid, v_hid, s_gz, v_gz, true);
      flood_layer(L_r_h, s_hid, v_hid, s_gr, v_gr, true);
    }
    sigmoid_kernel<<<(N * 64 + thr - 1) / thr, thr, 0, stream>>>(s_gz, N * 64);
    sigmoid_kernel<<<(N * 32 + thr - 1) / thr, thr, 0, stream>>>(v_gz, N * 32);
    sigmoid_kernel<<<(N * 64 + thr - 1) / thr, thr, 0, stream>>>(s_gr, N * 64);
    sigmoid_kernel<<<(N * 32 + thr - 1) / thr, thr, 0, stream>>>(v_gr, N * 32);
    flood_layer(L_hh_in, s_enc, v_enc, s_hh, v_hh, false);
    if (hid) {
      mul_kernel<<<(N * 64 + thr - 1) / thr, thr, 0, stream>>>(s_rt, s_gr,
                                                               s_hid, N * 64);
      mul_kernel<<<(N * 32 + thr - 1) / thr, thr, 0, stream>>>(v_rt, v_gr,
                                                               v_hid, N * 32);
      flood_layer(L_hh_h, s_rt, v_rt, s_hh, v_hh, true);
    }
    tanh_kernel<<<(N * 64 + thr - 1) / thr, thr, 0, stream>>>(s_hh, N * 64);
    tanh_kernel<<<(N * 32 + thr - 1) / thr, thr, 0, stream>>>(v_hh, N * 32);
    gru_kernel<<<(N * 64 + thr - 1) / thr, thr, 0, stream>>>(
        s_hid, s_gz, s_hh, seq, i - 1, T, 64, N, hid);
    gru_kernel<<<(N * 32 + thr - 1) / thr, thr, 0, stream>>>(
        v_hid, v_gz, v_hh, seq, i - 1, T, 32, N, hid);
    // feat_pred: g1 (acts) then g2 (no acts, SO=1, VO=2)
    gvp_node_kernel<64, 16, 64, 16, true, true, false>
        <<<nt16, 32, 0, stream>>>(s_hid, v_hid, g1.Wh, g1.Ws, g1.Wv, g1.Bs,
                                  s_f, v_f, N);
    gvp_node_kernel<64, 16, 1, 2, false, false, false>
        <<<nt16, 32, 0, stream>>>(s_f, v_f, g2.Wh, g2.Ws, g2.Wv, g2.Bs, l_pred,
                                  v_old, N);
    feat_epilogue_kernel<<<nb_n, thr, 0, stream>>>(
        l_pred, v_old, seq, binary, x_v, x_vn, rain, wdfp, l_h, s_h, v_h, r_h,
        (float*)d_out, scal, cnt, i, T, N, steps, out_size);
    loss_step_kernel<<<1, 1, 0, stream>>>(scal, cnt, N);
  }
  write_tail_kernel<<<(N * T + thr - 1) / thr, thr, 0, stream>>>(
      scal, seq, (float*)d_out, N, T, steps, out_size);
}